// RSGCNModel_37701222924447
// MI455X (gfx1250) — compile-verified
//
#include <hip/hip_runtime.h>
#include <math.h>

#define HID 16

typedef float v2f __attribute__((ext_vector_type(2)));
typedef float v8f __attribute__((ext_vector_type(8)));

__device__ __forceinline__ v8f wmma4(v2f a, v2f b, v8f c) {
  // V_WMMA_F32_16X16X4_F32 : D = A(16x4) * B(4x16) + C(16x16), all f32
  return __builtin_amdgcn_wmma_f32_16x16x4_f32(false, a, false, b, (short)0, c,
                                               false, false);
}

// ---------------- init: h0 from x (pad 14 -> 16), zero cnt ----------------
__global__ void k_init(const float* __restrict__ x, float* __restrict__ h0,
                       float* __restrict__ cnt, int N) {
  int gid = blockIdx.x * blockDim.x + threadIdx.x;
  if (gid < N * HID) {
    int n = gid >> 4, c = gid & 15;
    h0[gid] = (c < 14) ? x[n * 16 + 2 + c] : 0.0f;
  }
  if (gid < N) cnt[gid] = 0.0f;
}

// ---------------- per-dst edge count (same for all layers) ----------------
__global__ void k_count(const int* __restrict__ ei, float* __restrict__ cnt, int E) {
  int e = blockIdx.x * blockDim.x + threadIdx.x;
  if (e < E) atomicAdd(&cnt[ei[E + e]], 1.0f);
}

__global__ void k_zero(float* __restrict__ p, int n) {
  int gid = blockIdx.x * blockDim.x + threadIdx.x;
  if (gid < n) p[gid] = 0.0f;
}

// ------- edge message: msg[e,hf] = sum_c h[src,c]*relu(rel@Win+bin) -------
// 16 threads per edge (hf = output feature). Weight columns live in VGPRs
// (loop-invariant per thread), h[src] row broadcast via half-wave shuffle,
// ~90% of edges (cross-region) skip the einsum entirely.
template <int INC>
__global__ void k_edge_t(const float* __restrict__ x, const int* __restrict__ ei,
                         const int* __restrict__ region, const float* __restrict__ h,
                         const float* __restrict__ Wi, const float* __restrict__ bi,
                         float* __restrict__ agg, int E) {
  const int hf = threadIdx.x & 15;
  const int hbase = threadIdx.x & 16;  // base lane of this half-wave
  float w0[INC], w1[INC], bb[INC];
#pragma unroll
  for (int c = 0; c < INC; ++c) {
    w0[c] = Wi[c * HID + hf];
    w1[c] = Wi[INC * HID + c * HID + hf];
    bb[c] = bi[c * HID + hf];
  }
  const int stride = gridDim.x * blockDim.x;  // multiple of 16 -> hf invariant
  const int total = E * HID;
  for (int gid = blockIdx.x * blockDim.x + threadIdx.x; gid < total; gid += stride) {
    int e = gid >> 4;
    int s = ei[e];
    int d = ei[E + e];
    if (region[s] == region[d]) {  // uniform per half-wave
      float hval = h[s * HID + hf];
      float rx = x[d * 16 + 0] - x[s * 16 + 0];
      float ry = x[d * 16 + 1] - x[s * 16 + 1];
      float m = 0.0f;
#pragma unroll
      for (int c = 0; c < INC; ++c) {
        float sv = fmaf(rx, w0[c], fmaf(ry, w1[c], bb[c]));
        sv = fmaxf(sv, 0.0f);
        float hc = __shfl(hval, hbase + c, 32);
        m = fmaf(hc, sv, m);
      }
      atomicAdd(&agg[d * HID + hf], m);
    }
  }
}

// ------- node update: h' = relu((agg/max(cnt,1)) @ Wout + bout) via WMMA ---
// One wave per 16-node tile; K=16 -> 4x V_WMMA_F32_16X16X4_F32; bias as C.
__global__ void k_node(const float* __restrict__ agg, const float* __restrict__ cnt,
                       const float* __restrict__ Wo, const float* __restrict__ bo,
                       float* __restrict__ hout, int N) {
  int gwave = (blockIdx.x * blockDim.x + threadIdx.x) >> 5;
  int lane = threadIdx.x & 31;
  int half = lane >> 4;   // hi half-wave
  int lp = lane & 15;
  int base = gwave * 16;
  if (base >= N) return;  // wave-uniform
  int r = base + lp;
  int rc = (r < N) ? r : (N - 1);
  float scale = 1.0f / fmaxf(cnt[rc], 1.0f);
  v8f acc;
  float bv = bo[lp];
#pragma unroll
  for (int i = 0; i < 8; ++i) acc[i] = bv;
#pragma unroll
  for (int k = 0; k < 4; ++k) {
    // A layout: lane holds row (lane%16), cols {4k+2*half, 4k+2*half+1}
    const float2 av = *reinterpret_cast<const float2*>(agg + rc * HID + 4 * k + 2 * half);
    v2f a; a.x = av.x * scale; a.y = av.y * scale;
    v2f b;
    b.x = Wo[(4 * k + 2 * half + 0) * HID + lp];
    b.y = Wo[(4 * k + 2 * half + 1) * HID + lp];
    acc = wmma4(a, b, acc);
  }
  if (base + 16 <= N) {  // full tile: branch-free coalesced stores
#pragma unroll
    for (int i = 0; i < 8; ++i)
      hout[(base + i + 8 * half) * HID + lp] = fmaxf(acc[i], 0.0f);
  } else {
#pragma unroll
    for (int i = 0; i < 8; ++i) {
      int m = base + i + 8 * half;
      if (m < N) hout[m * HID + lp] = fmaxf(acc[i], 0.0f);
    }
  }
}

// ------- decoder: sigmoid(relu(relu([h_s|h_d]@Wd1+b1)@Wd2+b2)@Wd3+b3) -----
// One wave per 16-edge tile. 8+4+4 WMMAs; C->A re-layout through padded LDS.
__global__ void k_decoder(const float* __restrict__ h, const int* __restrict__ ei,
                          const float* __restrict__ Wd1, const float* __restrict__ bd1,
                          const float* __restrict__ Wd2, const float* __restrict__ bd2,
                          const float* __restrict__ Wd3, const float* __restrict__ bd3,
                          float* __restrict__ out, int E) {
  __shared__ float tile[8][16][17];  // 8 waves/block, +1 pad kills bank conflicts
  int widx = threadIdx.x >> 5;
  int gwave = (blockIdx.x * blockDim.x + threadIdx.x) >> 5;
  int lane = threadIdx.x & 31;
  int half = lane >> 4;
  int lp = lane & 15;
  int base = gwave * 16;
  if (base >= E) return;  // wave-uniform
  int e = base + lp; if (e >= E) e = E - 1;
  int s = ei[e], d = ei[E + e];

  // GEMM1: z(16x32) @ Wd1(32x16) + bd1 ; z row = [h[s] | h[d]]
  v8f acc;
  float bv = bd1[lp];
#pragma unroll
  for (int i = 0; i < 8; ++i) acc[i] = bv;
#pragma unroll
  for (int k = 0; k < 8; ++k) {
    int col = 4 * k + 2 * half;  // even, pair never straddles the 16 boundary
    const float* src = (col < 16) ? (h + s * HID + col) : (h + d * HID + (col - 16));
    const float2 av = *reinterpret_cast<const float2*>(src);
    v2f a; a.x = av.x; a.y = av.y;
    v2f b;
    b.x = Wd1[(col + 0) * HID + lp];
    b.y = Wd1[(col + 1) * HID + lp];
    acc = wmma4(a, b, acc);
  }
  // relu + C-layout -> LDS (row-major tile)
#pragma unroll
  for (int i = 0; i < 8; ++i) tile[widx][i + 8 * half][lp] = fmaxf(acc[i], 0.0f);
  __builtin_amdgcn_wave_barrier();

  // GEMM2: z1(16x16) @ Wd2(16x16) + bd2
  v8f acc2;
  bv = bd2[lp];
#pragma unroll
  for (int i = 0; i < 8; ++i) acc2[i] = bv;
#pragma unroll
  for (int k = 0; k < 4; ++k) {
    int col = 4 * k + 2 * half;
    v2f a;
    a.x = tile[widx][lp][col];
    a.y = tile[widx][lp][col + 1];
    v2f b;
    b.x = Wd2[(col + 0) * HID + lp];
    b.y = Wd2[(col + 1) * HID + lp];
    acc2 = wmma4(a, b, acc2);
  }
  __builtin_amdgcn_wave_barrier();
#pragma unroll
  for (int i = 0; i < 8; ++i) tile[widx][i + 8 * half][lp] = fmaxf(acc2[i], 0.0f);
  __builtin_amdgcn_wave_barrier();

  // GEMM3: z2(16x16) @ Wd3 packed into B column 0 (+ bd3 in C column 0)
  float bias3 = bd3[0];
  v8f acc3;
  float c3 = (lp == 0) ? bias3 : 0.0f;
#pragma unroll
  for (int i = 0; i < 8; ++i) acc3[i] = c3;
#pragma unroll
  for (int k = 0; k < 4; ++k) {
    int col = 4 * k + 2 * half;
    v2f a;
    a.x = tile[widx][lp][col];
    a.y = tile[widx][lp][col + 1];
    float w0 = Wd3[col + 0];
    float w1 = Wd3[col + 1];
    v2f b;
    b.x = (lp == 0) ? w0 : 0.0f;
    b.y = (lp == 0) ? w1 : 0.0f;
    acc3 = wmma4(a, b, acc3);
  }
  // D column 0 holds the logits: lanes with lp==0 store rows i + 8*half
  if (lp == 0) {
    if (base + 16 <= E) {
#pragma unroll
      for (int i = 0; i < 8; ++i)
        out[base + i + 8 * half] = 1.0f / (1.0f + __expf(-acc3[i]));
    } else {
#pragma unroll
      for (int i = 0; i < 8; ++i) {
        int m = base + i + 8 * half;
        if (m < E) out[m] = 1.0f / (1.0f + __expf(-acc3[i]));
      }
    }
  }
}

extern "C" void kernel_launch(void* const* d_in, const int* in_sizes, int n_in,
                              void* d_out, int out_size, void* d_ws, size_t ws_size,
                              hipStream_t stream) {
  const float* x      = (const float*)d_in[0];
  const int*   ei     = (const int*)d_in[1];
  const int*   region = (const int*)d_in[2];
  const float* Win[3]  = {(const float*)d_in[3],  (const float*)d_in[7],  (const float*)d_in[11]};
  const float* bin[3]  = {(const float*)d_in[4],  (const float*)d_in[8],  (const float*)d_in[12]};
  const float* Wout[3] = {(const float*)d_in[5],  (const float*)d_in[9],  (const float*)d_in[13]};
  const float* bout[3] = {(const float*)d_in[6],  (const float*)d_in[10], (const float*)d_in[14]};
  const float* Wd1 = (const float*)d_in[15];
  const float* bd1 = (const float*)d_in[16];
  const float* Wd2 = (const float*)d_in[17];
  const float* bd2 = (const float*)d_in[18];
  const float* Wd3 = (const float*)d_in[19];
  const float* bd3 = (const float*)d_in[20];

  const int N = in_sizes[2];
  const int E = in_sizes[1] / 2;

  float* ws  = (float*)d_ws;
  float* hA  = ws;                       // N*16
  float* hB  = ws + (size_t)N * 16;      // N*16
  float* agg = ws + (size_t)N * 32;      // N*16
  float* cnt = ws + (size_t)N * 48;      // N

  const int T = 256;
  const int nh  = N * HID;
  const int gNH = (nh + T - 1) / T;
  const int gE  = (E + T - 1) / T;
  // edge kernel: ~4 edges per thread via grid-stride (weights stay in VGPRs)
  const long long ethreads = ((long long)E * HID + 3) / 4;
  const int gEdge = (int)((ethreads + T - 1) / T);
  const int nwaves  = (N + 15) / 16;
  const int gNode   = (nwaves * 32 + T - 1) / T;
  const int ewaves  = (E + 15) / 16;
  const int gDec    = (ewaves * 32 + T - 1) / T;

  k_init<<<gNH, T, 0, stream>>>(x, hA, cnt, N);
  k_count<<<gE, T, 0, stream>>>(ei, cnt, E);

  float* hc = hA;
  float* hn = hB;
  for (int l = 0; l < 3; ++l) {
    k_zero<<<gNH, T, 0, stream>>>(agg, nh);
    if (l == 0)
      k_edge_t<14><<<gEdge, T, 0, stream>>>(x, ei, region, hc, Win[l], bin[l], agg, E);
    else
      k_edge_t<16><<<gEdge, T, 0, stream>>>(x, ei, region, hc, Win[l], bin[l], agg, E);
    k_node<<<gNode, T, 0, stream>>>(agg, cnt, Wout[l], bout[l], hn, N);
    float* tmp = hc; hc = hn; hn = tmp;
  }

  k_decoder<<<gDec, T, 0, stream>>>(hc, ei, Wd1, bd1, Wd2, bd2, Wd3, bd3,
                                    (float*)d_out, E);
}